// GATConv_23115513987274
// MI455X (gfx1250) — compile-verified
//
#include <hip/hip_runtime.h>

#define N_NODES 50000
#define N_EDGES 1600000
#define IN_SIZE 128
#define OUTH    128   // OUT_SIZE * HEADS
#define HEADS   4

typedef __attribute__((ext_vector_type(2))) float v2f;
typedef __attribute__((ext_vector_type(8))) float v8f;

// ---- order-preserving float <-> uint encoding for atomic max -------------
__device__ __forceinline__ unsigned enc_f32(float f) {
  unsigned u = __float_as_uint(f);
  return (u & 0x80000000u) ? ~u : (u | 0x80000000u);
}
__device__ __forceinline__ float dec_f32(unsigned e) {
  unsigned u = (e & 0x80000000u) ? (e ^ 0x80000000u) : ~e;
  return __uint_as_float(u);
}

// ---- Kernel 1: Zp = Z @ W^T + b via V_WMMA_F32_16X16X4_F32 ----------------
// One wave per 16x16 output tile. Block = 256 threads = 8 waves = 8 col tiles
// of one 16-node tile. 50000/16 = 3125 node tiles exactly.
__global__ __launch_bounds__(256) void gat_gemm_wmma(
    const float* __restrict__ Z, const float* __restrict__ W,
    const float* __restrict__ bias, float* __restrict__ Zp) {
  const int wave = threadIdx.x >> 5;   // 0..7 : column tile
  const int lane = threadIdx.x & 31;
  const int row  = lane & 15;
  const int kh   = lane >> 4;          // 0/1 : which K pair this lane holds
  const int nodeTile = blockIdx.x;

  const float* zrow = Z + (size_t)(nodeTile * 16 + row) * IN_SIZE; // A: M=row
  const float* wrow = W + (size_t)(wave * 16 + row) * IN_SIZE;     // B: N=row (B[k][n]=W[n][k])

  v8f acc = {};
  #pragma unroll 4
  for (int k = 0; k < IN_SIZE; k += 4) {
    v2f a, b;
    // A 16x4 f32 layout: VGPR0 = K0 (lanes 0-15) / K2 (lanes 16-31), VGPR1 = K1/K3
    a.x = zrow[k + 2 * kh];
    a.y = zrow[k + 2 * kh + 1];
    b.x = wrow[k + 2 * kh];
    b.y = wrow[k + 2 * kh + 1];
    acc = __builtin_amdgcn_wmma_f32_16x16x4_f32(
        /*neg_a=*/false, a, /*neg_b=*/false, b,
        /*c_mod=*/(short)0, acc, /*reuse_a=*/false, /*reuse_b=*/false);
  }

  // D layout: VGPR r -> M = r + 8*(lane/16), N = lane%16
  const int ncol  = wave * 16 + (lane & 15);
  const int group = lane >> 4;
  const float bv  = bias[ncol];
  float* out = Zp + (size_t)(nodeTile * 16) * OUTH + ncol;
  #pragma unroll
  for (int r = 0; r < 8; ++r) {
    out[(size_t)(r + 8 * group) * OUTH] = acc[r] + bv;
  }
}

// ---- Kernel 2: per-node attention logits e_l/e_r [N,4] --------------------
__global__ __launch_bounds__(256) void gat_logits(
    const float* __restrict__ Zp, const float* __restrict__ a_l,
    const float* __restrict__ a_r, float* __restrict__ e_l,
    float* __restrict__ e_r) {
  const int nd = blockIdx.x * blockDim.x + threadIdx.x;
  if (nd >= N_NODES) return;
  const float* z = Zp + (size_t)nd * OUTH;
  float accL0 = 0.f, accL1 = 0.f, accL2 = 0.f, accL3 = 0.f;
  float accR0 = 0.f, accR1 = 0.f, accR2 = 0.f, accR3 = 0.f;
  #pragma unroll 8
  for (int c = 0; c < OUTH; c += 4) {      // c%4 == head
    const float4 zv = *(const float4*)(z + c);
    const float4 al = *(const float4*)(a_l + c);
    const float4 ar = *(const float4*)(a_r + c);
    accL0 += zv.x * al.x; accL1 += zv.y * al.y;
    accL2 += zv.z * al.z; accL3 += zv.w * al.w;
    accR0 += zv.x * ar.x; accR1 += zv.y * ar.y;
    accR2 += zv.z * ar.z; accR3 += zv.w * ar.w;
  }
  float* pl = e_l + (size_t)nd * 4;
  float* pr = e_r + (size_t)nd * 4;
  pl[0] = accL0; pl[1] = accL1; pl[2] = accL2; pl[3] = accL3;
  pr[0] = accR0; pr[1] = accR1; pr[2] = accR2; pr[3] = accR3;
}

__device__ __forceinline__ void edge_logits(const float* e_l, const float* e_r,
                                            int src, int dst, float a[4]) {
  #pragma unroll
  for (int h = 0; h < HEADS; ++h) {
    float v = e_l[(size_t)src * 4 + h] + e_r[(size_t)dst * 4 + h];
    a[h] = v > 0.f ? v : 0.01f * v;  // leaky_relu, slope 0.01
  }
}

// ---- Kernel 3: segment max over dst (encoded-uint atomic max) -------------
__global__ __launch_bounds__(256) void gat_edge_max(
    const int* __restrict__ idx, const float* __restrict__ e_l,
    const float* __restrict__ e_r, unsigned* __restrict__ m_enc) {
  const int e = blockIdx.x * blockDim.x + threadIdx.x;
  if (e >= N_EDGES) return;
  const int src = idx[e];
  const int dst = idx[N_EDGES + e];
  float a[4];
  edge_logits(e_l, e_r, src, dst, a);
  #pragma unroll
  for (int h = 0; h < HEADS; ++h)
    atomicMax(&m_enc[(size_t)dst * 4 + h], enc_f32(a[h]));
}

// ---- Kernel 4: segment sum of exp(a - m) over dst -------------------------
__global__ __launch_bounds__(256) void gat_edge_sum(
    const int* __restrict__ idx, const float* __restrict__ e_l,
    const float* __restrict__ e_r, const unsigned* __restrict__ m_enc,
    float* __restrict__ s) {
  const int e = blockIdx.x * blockDim.x + threadIdx.x;
  if (e >= N_EDGES) return;
  const int src = idx[e];
  const int dst = idx[N_EDGES + e];
  float a[4];
  edge_logits(e_l, e_r, src, dst, a);
  #pragma unroll
  for (int h = 0; h < HEADS; ++h) {
    const float m = dec_f32(m_enc[(size_t)dst * 4 + h]);
    atomicAdd(&s[(size_t)dst * 4 + h], __expf(a[h] - m));
  }
}

// ---- Kernel 5: rst[src] += att * Zp[dst] ; one wave per edge --------------
// Lane L owns columns [4L, 4L+4): coalesced float4 gather of Zp[dst],
// 4 global f32 atomics into the L2-resident output.
__global__ __launch_bounds__(256) void gat_scatter(
    const int* __restrict__ idx, const float* __restrict__ Zp,
    const float* __restrict__ e_l, const float* __restrict__ e_r,
    const unsigned* __restrict__ m_enc, const float* __restrict__ s,
    float* __restrict__ rst) {
  const int lane = threadIdx.x & 31;
  const int e = blockIdx.x * 8 + (threadIdx.x >> 5);
  if (e >= N_EDGES) return;
  const int src = idx[e];
  const int dst = idx[N_EDGES + e];
  float a[4], att[4];
  edge_logits(e_l, e_r, src, dst, a);
  #pragma unroll
  for (int h = 0; h < HEADS; ++h) {
    const float m  = dec_f32(m_enc[(size_t)dst * 4 + h]);
    const float ex = __expf(a[h] - m);
    att[h] = ex / s[(size_t)dst * 4 + h];
  }
  const float4 zv = *(const float4*)(Zp + (size_t)dst * OUTH + lane * 4);
  float* o = rst + (size_t)src * OUTH + lane * 4;
  atomicAdd(o + 0, att[0] * zv.x);
  atomicAdd(o + 1, att[1] * zv.y);
  atomicAdd(o + 2, att[2] * zv.z);
  atomicAdd(o + 3, att[3] * zv.w);
}

extern "C" void kernel_launch(void* const* d_in, const int* in_sizes, int n_in,
                              void* d_out, int out_size, void* d_ws, size_t ws_size,
                              hipStream_t stream) {
  const int*   idx = (const int*)d_in[0];     // [2, E]
  // d_in[1] = n (scalar), compile-time constant here
  const float* Z   = (const float*)d_in[2];   // [N, 128]
  const float* W   = (const float*)d_in[3];   // [128, 128]
  const float* b   = (const float*)d_in[4];   // [128]
  const float* a_l = (const float*)d_in[5];   // [1, 32, 4] -> flat 128
  const float* a_r = (const float*)d_in[6];
  float* rst = (float*)d_out;                 // [N, 32, 4] flat

  // workspace carve-up
  char* ws = (char*)d_ws;
  const size_t zp_bytes = (size_t)N_NODES * OUTH * sizeof(float); // 25.6 MB
  const size_t nh_bytes = (size_t)N_NODES * HEADS * sizeof(float); // 800 KB
  float*    Zp    = (float*)(ws);
  float*    e_l   = (float*)(ws + zp_bytes);
  float*    e_r   = (float*)(ws + zp_bytes + nh_bytes);
  unsigned* m_enc = (unsigned*)(ws + zp_bytes + 2 * nh_bytes);
  float*    ssum  = (float*)(ws + zp_bytes + 3 * nh_bytes);

  // per-call re-init (graph-capture-safe stream memsets)
  hipMemsetAsync(rst,   0, zp_bytes, stream);   // output accumulators
  hipMemsetAsync(m_enc, 0, nh_bytes, stream);   // enc(0)=-inf-equivalent floor
  hipMemsetAsync(ssum,  0, nh_bytes, stream);

  gat_gemm_wmma<<<N_NODES / 16, 256, 0, stream>>>(Z, W, b, Zp);
  gat_logits<<<(N_NODES + 255) / 256, 256, 0, stream>>>(Zp, a_l, a_r, e_l, e_r);
  gat_edge_max<<<(N_EDGES + 255) / 256, 256, 0, stream>>>(idx, e_l, e_r, m_enc);
  gat_edge_sum<<<(N_EDGES + 255) / 256, 256, 0, stream>>>(idx, e_l, e_r, m_enc, ssum);
  gat_scatter<<<N_EDGES / 8, 256, 0, stream>>>(idx, Zp, e_l, e_r, m_enc, ssum, rst);
}